// AgMixPooler_10548439679365
// MI455X (gfx1250) — compile-verified
//
#include <hip/hip_runtime.h>
#include <math.h>

typedef __attribute__((ext_vector_type(2))) float v2f;
typedef __attribute__((ext_vector_type(8))) float v8f;

#define B_     16
#define T_     4096
#define E_     1024
#define LOUT_  512
#define BT_    (B_ * T_)
#define POOLED_ ((long)B_ * LOUT_ * E_)

// workspace layout (in floats)
#define CWP_OFF 0                       // 16*1024 floats (zero-padded conv_w)
#define P_OFF   (16 * 1024)             // BT_*8 floats (projection p)
#define A_OFF   (P_OFF + BT_ * 8)       // BT_ floats (pre-softmax logits)

// ---------------------------------------------------------------------------
// Kernel 0: zero-pad conv_w (7,1024) -> cwp (16,1024) so WMMA B-fragment loads
// need no predication (EXEC must be all-1s around v_wmma).
// ---------------------------------------------------------------------------
__global__ __launch_bounds__(256)
void pad_convw_kernel(const float* __restrict__ conv_w, float* __restrict__ cwp) {
    int idx = blockIdx.x * 256 + threadIdx.x;          // 0 .. 16*1024-1
    int row = idx >> 10;
    cwp[idx] = (row < 7) ? conv_w[idx] : 0.0f;
}

// ---------------------------------------------------------------------------
// Kernel 1: p = X @ cwp^T via V_WMMA_F32_16X16X4_F32.
// One wave per 16-row tile of X (BT/16 = 4096 tiles). K=1024 in 16 chunks of
// 64, staged through LDS (row padded to 68 floats: 16B-aligned float4 stores,
// conflict-free A-fragment reads).
// ---------------------------------------------------------------------------
__global__ __launch_bounds__(32)
void conv_proj_wmma_kernel(const float* __restrict__ x,
                           const float* __restrict__ cwp,
                           float* __restrict__ p) {
    __shared__ __align__(16) float lds[16 * 68];

    const int lane = threadIdx.x;        // 0..31
    const int half = lane >> 4;          // 0: K+0..1, 1: K+2..3
    const int l16  = lane & 15;
    const long row0 = (long)blockIdx.x * 16;

    v8f acc = {};                        // 16x16 f32 accumulator (8 VGPRs)

    for (int kc = 0; kc < 16; ++kc) {
        const int kbase = kc * 64;

        // Stage 16x64 chunk: half-wave-coalesced float4 loads, 2 rows/iter.
        #pragma unroll
        for (int rr = 0; rr < 8; ++rr) {
            const int r = rr * 2 + half;
            const float4 v = *(const float4*)(x + (row0 + r) * E_ + kbase + l16 * 4);
            *(float4*)(&lds[r * 68 + l16 * 4]) = v;
        }
        __syncthreads();

        #pragma unroll
        for (int kk = 0; kk < 16; ++kk) {
            const int ko = kk * 4 + half * 2;
            v2f a, b;
            // A fragment: M = l16, K = ko..ko+1 (ISA 16x4 f32 layout)
            a.x = lds[l16 * 68 + ko];
            a.y = lds[l16 * 68 + ko + 1];
            // B fragment: N = l16 (conv tap, rows >=7 are zero), K = ko..ko+1
            const float* bp = cwp + l16 * E_ + kbase + ko;
            b.x = bp[0];
            b.y = bp[1];
            acc = __builtin_amdgcn_wmma_f32_16x16x4_f32(
                false, a, false, b, (short)0, acc, false, false);
        }
        __syncthreads();
    }

    // D layout: VGPR r -> M = r + 8*half, N = l16. Only taps 0..6 matter
    // (col 7 is exactly zero); store cols 0..7 for aligned stride-8 rows.
    if (l16 < 8) {
        #pragma unroll
        for (int r = 0; r < 8; ++r) {
            const long row = row0 + r + half * 8;
            p[row * 8 + l16] = acc[r];
        }
    }
}

// ---------------------------------------------------------------------------
// Kernel 2: combine shifted diagonals of p + ssf projection, gate, tanh, mask.
// ---------------------------------------------------------------------------
__global__ __launch_bounds__(256)
void combine_kernel(const float* __restrict__ p,
                    const float* __restrict__ ssf_x,
                    const unsigned char* __restrict__ mask,
                    const float* __restrict__ conv_b,
                    const float* __restrict__ ssf_weight,
                    const float* __restrict__ ssf_bias,
                    const float* __restrict__ gate_logit,
                    float* __restrict__ a_out) {
    const int g = blockIdx.x * 256 + threadIdx.x;      // b*T + t
    if (g >= BT_) return;
    const int b = g / T_;
    const int t = g - b * T_;

    float wconv = conv_b[0];
    #pragma unroll
    for (int kh = 0; kh < 7; ++kh) {
        const int tau = t + kh - 3;
        if (tau >= 0 && tau < T_)
            wconv += p[((long)b * T_ + tau) * 8 + kh];
    }

    float wssf = ssf_bias[0];
    #pragma unroll
    for (int i = 0; i < 7; ++i)
        wssf += ssf_x[(long)g * 7 + i] * ssf_weight[i];

    const float alpha = 1.0f / (1.0f + expf(-gate_logit[0]));
    const float av = tanhf(alpha * wconv + (1.0f - alpha) * wssf);
    a_out[g] = mask[g] ? av : -INFINITY;
}

// ---------------------------------------------------------------------------
// Kernel 3: per-batch softmax over T=4096. One 256-thread block per batch.
// ---------------------------------------------------------------------------
__global__ __launch_bounds__(256)
void softmax_kernel(const float* __restrict__ a, float* __restrict__ attn) {
    __shared__ float red[256];
    const int b = blockIdx.x;
    const int tid = threadIdx.x;
    const float* ab = a + (long)b * T_;

    float m = -INFINITY;
    for (int t = tid; t < T_; t += 256) m = fmaxf(m, ab[t]);
    red[tid] = m; __syncthreads();
    for (int s = 128; s > 0; s >>= 1) {
        if (tid < s) red[tid] = fmaxf(red[tid], red[tid + s]);
        __syncthreads();
    }
    m = red[0]; __syncthreads();

    float sum = 0.0f;
    for (int t = tid; t < T_; t += 256) {
        const float v = ab[t];
        sum += (v == -INFINITY) ? 0.0f : expf(v - m);
    }
    red[tid] = sum; __syncthreads();
    for (int s = 128; s > 0; s >>= 1) {
        if (tid < s) red[tid] += red[tid + s];
        __syncthreads();
    }
    const float inv = 1.0f / red[0];

    for (int t = tid; t < T_; t += 256) {
        const float v = ab[t];
        attn[(long)b * T_ + t] = (v == -INFINITY) ? 0.0f : expf(v - m) * inv;
    }
}

// ---------------------------------------------------------------------------
// Kernel 4: segment pooling. seg(t) = t/8; one block per (b, s); each thread
// owns 4 consecutive e-columns (float4), 8-tap weighted sum, normalize.
// ---------------------------------------------------------------------------
__global__ __launch_bounds__(256)
void pool_kernel(const float* __restrict__ x,
                 const float* __restrict__ attn,
                 const unsigned char* __restrict__ mask,
                 float* __restrict__ pooled) {
    const int blk = blockIdx.x;          // b*LOUT + s
    const int b = blk / LOUT_;
    const int s = blk - b * LOUT_;
    const int t0 = s * (T_ / LOUT_);     // segment = 8 consecutive t

    float w[8];
    float sw = 0.0f;
    #pragma unroll
    for (int i = 0; i < 8; ++i) {
        const long g = (long)b * T_ + t0 + i;
        const float wi = attn[g] * (mask[g] ? 1.0f : 0.0f);
        w[i] = wi;
        sw += wi;
    }
    const float inv = 1.0f / fmaxf(sw, 1e-8f);

    const int e = threadIdx.x * 4;
    float4 acc = {0.0f, 0.0f, 0.0f, 0.0f};
    #pragma unroll
    for (int i = 0; i < 8; ++i) {
        const float4 v = *(const float4*)(x + ((long)b * T_ + t0 + i) * E_ + e);
        acc.x += w[i] * v.x; acc.y += w[i] * v.y;
        acc.z += w[i] * v.z; acc.w += w[i] * v.w;
    }
    float4 o = {acc.x * inv, acc.y * inv, acc.z * inv, acc.w * inv};
    *(float4*)(pooled + ((long)b * LOUT_ + s) * E_ + e) = o;
}

// ---------------------------------------------------------------------------
extern "C" void kernel_launch(void* const* d_in, const int* in_sizes, int n_in,
                              void* d_out, int out_size, void* d_ws, size_t ws_size,
                              hipStream_t stream) {
    const float* x          = (const float*)d_in[0];          // (B,T,E)
    const float* ssf_x      = (const float*)d_in[1];          // (B,T,7)
    const unsigned char* pm = (const unsigned char*)d_in[2];  // (B,T) bool
    const float* conv_w     = (const float*)d_in[3];          // (1,1,7,E)
    const float* conv_b     = (const float*)d_in[4];          // (1,)
    const float* ssf_w      = (const float*)d_in[5];          // (7,)
    const float* ssf_b      = (const float*)d_in[6];          // (1,)
    const float* gate       = (const float*)d_in[7];          // (1,)

    float* ws   = (float*)d_ws;
    float* cwp  = ws + CWP_OFF;
    float* p    = ws + P_OFF;
    float* a    = ws + A_OFF;

    float* pooled = (float*)d_out;               // (B, L_OUT, E)
    float* attn   = pooled + POOLED_;            // (B, T, 1)

    pad_convw_kernel<<<(16 * 1024) / 256, 256, 0, stream>>>(conv_w, cwp);
    conv_proj_wmma_kernel<<<BT_ / 16, 32, 0, stream>>>(x, cwp, p);
    combine_kernel<<<BT_ / 256, 256, 0, stream>>>(p, ssf_x, pm, conv_b, ssf_w,
                                                  ssf_b, gate, a);
    softmax_kernel<<<B_, 256, 0, stream>>>(a, attn);
    pool_kernel<<<B_ * LOUT_, 256, 0, stream>>>(x, attn, pm, pooled);
}